// LambdaRankLoss_5695126634556
// MI455X (gfx1250) — compile-verified
//
#include <hip/hip_runtime.h>
#include <hip/hip_bf16.h>
#include <cstdint>

// ---------------------------------------------------------------------------
// LambdaRank loss, N=4096.
// K1: single 1024-thread workgroup: bitonic sorts in LDS + maxDCG + pack[]
// K2: 16x16 grid of 256-thread blocks, each does a 256x256 pair tile.
//     Column tile staged into LDS with global_load_async_to_lds_b128 (CDNA5
//     async path, ASYNCcnt), block reduction finished with
//     V_WMMA_F32_16X16X4_F32 (exact f32 row-sum trick: A x ones + C).
// K3: one wave reduces the 256 block partials with the same WMMA trick.
// ---------------------------------------------------------------------------

#define SORT_T 1024
#define MAXN   4096
#define TILE   256
#define LOG2EPS (-33.2192809489f)   // log2(1e-10)

typedef __attribute__((ext_vector_type(2))) float v2f;
typedef __attribute__((ext_vector_type(8))) float v8f;

// Sum 256 f32 values (8 per lane of one wave32) exactly, using
// V_WMMA_F32_16X16X4_F32: D = A(16x4) * ones(4x16) + C replicates row sums
// across columns; column 0 lives in lanes 0 (M=0..7) and 16 (M=8..15).
__device__ __forceinline__ float wave_sum_256(const float v[8]) {
#if __has_builtin(__builtin_amdgcn_wmma_f32_16x16x4_f32)
  v8f c = {0.f, 0.f, 0.f, 0.f, 0.f, 0.f, 0.f, 0.f};
  v2f ones; ones[0] = 1.f; ones[1] = 1.f;
#pragma unroll
  for (int q = 0; q < 4; ++q) {
    v2f a; a[0] = v[2 * q]; a[1] = v[2 * q + 1];
    c = __builtin_amdgcn_wmma_f32_16x16x4_f32(
        /*neg_a=*/false, a, /*neg_b=*/false, ones,
        /*c_mod=*/(short)0, c, /*reuse_a=*/false, /*reuse_b=*/false);
  }
  float t = 0.f;
#pragma unroll
  for (int q = 0; q < 8; ++q) t += c[q];
  return __shfl(t, 0, 32) + __shfl(t, 16, 32);
#else
  // Fallback: plain wave32 butterfly reduction (ds_swizzle/permute path).
  float t = 0.f;
#pragma unroll
  for (int q = 0; q < 8; ++q) t += v[q];
#pragma unroll
  for (int off = 16; off > 0; off >>= 1) t += __shfl_xor(t, off, 32);
  return t;
#endif
}

// ---------------------------------------------------------------------------
// Kernel 1: sorts + per-item quantities. One workgroup, 1024 threads, LDS:
// 3 x 16KB arrays + 4KB reduction = 52KB (well under 320KB/WGP).
// pack[i] = { s_sorted_i, t_sorted_i, G_i, invD_i }
// ---------------------------------------------------------------------------
__global__ __launch_bounds__(SORT_T) void lrank_sort_prep(
    const float* __restrict__ preds, const float* __restrict__ labels,
    const int* __restrict__ kptr, float4* __restrict__ pack, int n) {
  __shared__ float k1[MAXN];   // pred keys  (sorted descending)
  __shared__ float v1[MAXN];   // labels carried with pred sort
  __shared__ float k2[MAXN];   // labels sorted descending (independent)
  __shared__ float red[SORT_T];

  const int tid = threadIdx.x;
  const int kval = *kptr;

  for (int e = tid; e < n; e += SORT_T) {
    float p = preds[e], l = labels[e];
    k1[e] = p; v1[e] = l; k2[e] = l;
  }

  // Bitonic sort, both sequences fused per compare step (descending).
  for (unsigned k = 2; k <= (unsigned)n; k <<= 1) {
    for (unsigned j = k >> 1; j > 0; j >>= 1) {
      __syncthreads();
      for (unsigned e = tid; e < (unsigned)n; e += SORT_T) {
        unsigned x = e ^ j;
        if (x > e) {
          const bool maxAtE = ((e & k) == 0);   // descending overall
          float a = k1[e], b = k1[x];
          if (maxAtE ? (a < b) : (a > b)) {
            k1[e] = b; k1[x] = a;
            float t = v1[e]; v1[e] = v1[x]; v1[x] = t;
          }
          float c = k2[e], d = k2[x];
          if (maxAtE ? (c < d) : (c > d)) { k2[e] = d; k2[x] = c; }
        }
      }
    }
  }
  __syncthreads();

  // maxDCG = max(sum_{j<k} (2^max(yt_j,0)-1)/log2(j+2), eps)
  float p = 0.f;
  for (int e = tid; e < n; e += SORT_T)
    if (e < kval)
      p += (exp2f(fmaxf(k2[e], 0.f)) - 1.f) * (1.f / __log2f((float)(e + 2)));
  red[tid] = p;
  __syncthreads();
  for (int s = SORT_T / 2; s > 0; s >>= 1) {
    if (tid < s) red[tid] += red[tid + s];
    __syncthreads();
  }
  const float rMax = 1.f / fmaxf(red[0], 1e-10f);

  for (int e = tid; e < n; e += SORT_T) {
    float invD = 1.f / __log2f((float)(e + 2));
    float G = (exp2f(fmaxf(v1[e], 0.f)) - 1.f) * rMax;
    pack[e] = make_float4(k1[e], v1[e], G, invD);
  }
}

// ---------------------------------------------------------------------------
// Kernel 2: 256x256 pair tile per block. Column tile staged via CDNA5 async
// DMA to LDS; broadcast ds_load_b128 reads in the inner loop; block sum via
// WMMA reduction; one partial per block (deterministic, no float atomics).
// ---------------------------------------------------------------------------
__global__ __launch_bounds__(TILE) void lrank_pairs(
    const float4* __restrict__ pack, const int* __restrict__ kptr,
    float* __restrict__ partials, int n) {
  __shared__ float4 tile[TILE];   // 4KB column tile {s,t,G,invD}
  __shared__ float  red[TILE];

  const int tid = threadIdx.x;
  const int i   = blockIdx.x * TILE + tid;   // this thread's row
  const int jc0 = blockIdx.y * TILE;         // column tile base
  const int kval = *kptr;

  // Async-stage the column tile straight into LDS (one b128 per lane).
  {
    unsigned long long ga = (unsigned long long)(const void*)(pack + (jc0 + tid));
    unsigned ld = (unsigned)(size_t)&tile[tid];   // low 32 bits = LDS offset
    asm volatile("global_load_async_to_lds_b128 %0, %1, off"
                 :: "v"(ld), "v"(ga) : "memory");
    asm volatile("s_wait_asynccnt 0" ::: "memory");
  }
  __syncthreads();

  const float4 r = pack[i];
  const float si = r.x, ti = r.y, Gi = r.z, Di = r.w;
  const bool rowOk = (i < kval);

  float acc = 0.f;
#pragma unroll 4
  for (int j = 0; j < TILE; ++j) {
    const float4 cj = tile[j];
    float d = si - cj.x;
    d = fminf(fmaxf(d, -1e8f), 1e8f);
    // log2(max(sigmoid(d), eps)); exp overflow -> sig=0 -> clamped at eps
    float sig = 1.f / (1.f + __expf(-d));
    float lg  = __log2f(fmaxf(sig, 1e-10f));
    float w   = fabsf(Di - cj.w) * fabsf(Gi - cj.z);
    float l   = fmaxf(w * lg, LOG2EPS);
    bool  m   = (ti > cj.y) && rowOk && ((jc0 + j) < kval);
    acc += m ? l : 0.f;
  }

  red[tid] = acc;
  __syncthreads();
  if (tid < 32) {                 // wave 0, EXEC all ones -> WMMA legal
    float v[8];
#pragma unroll
    for (int q = 0; q < 8; ++q) v[q] = red[tid + 32 * q];
    float s = wave_sum_256(v);
    if (tid == 0) partials[blockIdx.y * gridDim.x + blockIdx.x] = s;
  }
}

// ---------------------------------------------------------------------------
// Kernel 3: one wave reduces the block partials (nb <= 256), writes -sum.
// ---------------------------------------------------------------------------
__global__ __launch_bounds__(32) void lrank_final(
    const float* __restrict__ partials, float* __restrict__ out, int nb) {
  const int lane = threadIdx.x;
  float v[8];
#pragma unroll
  for (int q = 0; q < 8; ++q) {
    int idx = lane + 32 * q;
    v[q] = (idx < nb) ? partials[idx] : 0.f;
  }
  float t = wave_sum_256(v);
  if (lane == 0) out[0] = -t;
}

// ---------------------------------------------------------------------------
extern "C" void kernel_launch(void* const* d_in, const int* in_sizes, int n_in,
                              void* d_out, int out_size, void* d_ws, size_t ws_size,
                              hipStream_t stream) {
  const float* preds  = (const float*)d_in[0];
  const float* labels = (const float*)d_in[1];
  const int*   kptr   = (const int*)d_in[2];
  const int n = in_sizes[0];            // 4096 (power of two, multiple of 256)

  float4* pack     = (float4*)d_ws;
  float*  partials = (float*)((char*)d_ws + (size_t)n * sizeof(float4));

  lrank_sort_prep<<<1, SORT_T, 0, stream>>>(preds, labels, kptr, pack, n);

  const int tiles = n / TILE;           // 16
  dim3 grid(tiles, tiles);              // 256 blocks x 8 waves = 2048 waves
  lrank_pairs<<<grid, TILE, 0, stream>>>(pack, kptr, partials, n);

  lrank_final<<<1, 32, 0, stream>>>(partials, (float*)d_out, tiles * tiles);
}